// SAGE_3204045603665
// MI455X (gfx1250) — compile-verified
//
#include <hip/hip_runtime.h>

#define N_NODES 100000
#define N_EDGES 600000
#define D_HID   128

typedef float v2f __attribute__((ext_vector_type(2)));
typedef float v8f __attribute__((ext_vector_type(8)));

// ---------------------------------------------------------------------------
// In-degree: deg[d] = # edges with dst == d (same graph for all 3 layers)
// ---------------------------------------------------------------------------
__global__ void sage_degree_kernel(const int* __restrict__ dst,
                                   float* __restrict__ deg, int nE) {
    int e = blockIdx.x * blockDim.x + threadIdx.x;
    if (e < nE) atomicAdd(&deg[dst[e]], 1.0f);
}

// ---------------------------------------------------------------------------
// Scatter-add: agg[dst[e], :] += h[src[e], :]   (d = 128)
// One wave (32 lanes) per edge; each lane moves a float4 (coalesced B128 load,
// 4 f32 global atomics that resolve in L2 — the 51MB agg fits in 192MB L2).
// ---------------------------------------------------------------------------
__global__ void sage_scatter_kernel(const float* __restrict__ h,
                                    const int* __restrict__ src,
                                    const int* __restrict__ dst,
                                    float* __restrict__ agg, int nE) {
    long long gid = (long long)blockIdx.x * blockDim.x + threadIdx.x;
    int e    = (int)(gid >> 5);
    int lane = (int)(gid & 31);
    if (e >= nE) return;
    int s = src[e], d = dst[e];
    float4 v = ((const float4*)(h + (size_t)s * D_HID))[lane];
    float* o = agg + (size_t)d * D_HID + lane * 4;
    atomicAdd(o + 0, v.x);
    atomicAdd(o + 1, v.y);
    atomicAdd(o + 2, v.z);
    atomicAdd(o + 3, v.w);
}

// ---------------------------------------------------------------------------
// Fused SAGE layer GEMM: OUT = act(H @ Wself + (AGG/deg) @ Wneigh + bias)
// One wave per 16x16 output tile. K = 128, stepped 4 at a time with
// V_WMMA_F32_16X16X4_F32 (full f32 precision, matches the f32 reference).
// DIMN is a compile-time constant (128 or 64) so all W/OUT row strides fold
// into the 24-bit immediate offsets of global_load/global_store.
//
// f32 WMMA VGPR layouts (ISA 7.12.2):
//   A 16x4:  lanes 0-15 row=lr hold K={k,k+1} in {v0,v1}; lanes 16-31 K={k+2,k+3}
//   B 4x16:  lanes 0-15 col=lr hold K-rows {k,k+1}; lanes 16-31 rows {k+2,k+3}
//   C 16x16: VGPR i: lanes 0-15 -> (M=i, N=lr), lanes 16-31 -> (M=i+8, N=lr)
// ---------------------------------------------------------------------------
template <int RELU, int DIMN>
__global__ void sage_wmma_layer_kernel(const float* __restrict__ H,
                                       const float* __restrict__ AGG,
                                       const float* __restrict__ deg,
                                       const float* __restrict__ Wself,
                                       const float* __restrict__ Wneigh,
                                       const float* __restrict__ bias,
                                       float* __restrict__ OUT) {
    const int K    = D_HID;
    int tr   = blockIdx.x;            // 16-row node tile
    int tc   = blockIdx.y;            // 16-col output tile
    int lane = threadIdx.x;           // 0..31, full wave (N_NODES % 16 == 0)
    int half = lane >> 4;             // 0: K lo pair, 1: K hi pair
    int lr   = lane & 15;

    int arow = tr * 16 + lr;          // node row this lane feeds into A
    int bcol = tc * 16 + lr;          // output column this lane holds in B/C

    const float* hrow = H   + (size_t)arow * K;
    const float* grow = AGG + (size_t)arow * K;
    const float* ws   = Wself  + bcol;   // column base; row stride = DIMN (const)
    const float* wn   = Wneigh + bcol;
    float inv = 1.0f / fmaxf(deg[arow], 1.0f);   // mean aggregation

    v8f acc = {};
    #pragma unroll 4
    for (int k = 0; k < K; k += 4) {
        int ka = k + 2 * half;
        v2f a_s, a_n, b_s, b_n;
        a_s.x = hrow[ka];
        a_s.y = hrow[ka + 1];
        a_n.x = grow[ka] * inv;
        a_n.y = grow[ka + 1] * inv;
        b_s.x = ws[ka * DIMN];
        b_s.y = ws[(ka + 1) * DIMN];
        b_n.x = wn[ka * DIMN];
        b_n.y = wn[(ka + 1) * DIMN];
        // D = A*B + C, chained: self then neigh into the same accumulator
        acc = __builtin_amdgcn_wmma_f32_16x16x4_f32(
            false, a_s, false, b_s, (short)0, acc, false, false);
        acc = __builtin_amdgcn_wmma_f32_16x16x4_f32(
            false, a_n, false, b_n, (short)0, acc, false, false);
    }

    float bb = bias[bcol];
    float* orow = OUT + (size_t)(tr * 16 + 8 * half) * DIMN + bcol;
    #pragma unroll
    for (int i = 0; i < 8; ++i) {
        float v = acc[i] + bb;
        if (RELU) v = fmaxf(v, 0.0f);
        orow[i * DIMN] = v;          // row stride constant -> immediate offsets
    }
}

// ---------------------------------------------------------------------------
extern "C" void kernel_launch(void* const* d_in, const int* in_sizes, int n_in,
                              void* d_out, int out_size, void* d_ws, size_t ws_size,
                              hipStream_t stream) {
    (void)in_sizes; (void)n_in; (void)out_size; (void)ws_size;

    const float* feat = (const float*)d_in[0];
    const int*   src  = (const int*)  d_in[1];
    const int*   dst  = (const int*)  d_in[2];
    const float* Ws0  = (const float*)d_in[3];
    const float* Wn0  = (const float*)d_in[4];
    const float* b0   = (const float*)d_in[5];
    const float* Ws1  = (const float*)d_in[6];
    const float* Wn1  = (const float*)d_in[7];
    const float* b1   = (const float*)d_in[8];
    const float* Ws2  = (const float*)d_in[9];
    const float* Wn2  = (const float*)d_in[10];
    const float* b2   = (const float*)d_in[11];
    float*       out  = (float*)d_out;

    // Workspace layout (all f32): deg[N] | agg[N*128] | h1[N*128] | h2[N*128]
    float* deg = (float*)d_ws;
    float* agg = deg + N_NODES;
    float* h1  = agg + (size_t)N_NODES * D_HID;
    float* h2  = h1  + (size_t)N_NODES * D_HID;

    const size_t aggBytes = (size_t)N_NODES * D_HID * sizeof(float);
    const int scatterBlocks = (int)(((long long)N_EDGES * 32 + 255) / 256);
    dim3 gHid(N_NODES / 16, D_HID / 16);   // 6250 x 8
    dim3 gOut(N_NODES / 16, 64 / 16);      // 6250 x 4

    // degree (once) + layer 0 aggregation
    hipMemsetAsync(deg, 0, N_NODES * sizeof(float), stream);
    hipMemsetAsync(agg, 0, aggBytes, stream);
    sage_degree_kernel<<<(N_EDGES + 255) / 256, 256, 0, stream>>>(dst, deg, N_EDGES);
    sage_scatter_kernel<<<scatterBlocks, 256, 0, stream>>>(feat, src, dst, agg, N_EDGES);
    sage_wmma_layer_kernel<1, D_HID><<<gHid, 32, 0, stream>>>(feat, agg, deg, Ws0, Wn0, b0, h1);

    // layer 1
    hipMemsetAsync(agg, 0, aggBytes, stream);
    sage_scatter_kernel<<<scatterBlocks, 256, 0, stream>>>(h1, src, dst, agg, N_EDGES);
    sage_wmma_layer_kernel<1, D_HID><<<gHid, 32, 0, stream>>>(h1, agg, deg, Ws1, Wn1, b1, h2);

    // layer 2 (d_out = 64, no activation) -> d_out
    hipMemsetAsync(agg, 0, aggBytes, stream);
    sage_scatter_kernel<<<scatterBlocks, 256, 0, stream>>>(h2, src, dst, agg, N_EDGES);
    sage_wmma_layer_kernel<0, 64><<<gOut, 32, 0, stream>>>(h2, agg, deg, Ws2, Wn2, b2, out);
}